// Projector_5935644803461
// MI455X (gfx1250) — compile-verified
//
#include <hip/hip_runtime.h>

// ---------------------------------------------------------------------------
// Projector for MI455X (gfx1250, wave32).
//   out[b,0,i,j] = sum_k trilinear(vol, u0(b,i,j) + k * R2(b))
// u0 computed via V_WMMA_F32_16X16X4_F32 (the reference's einsum), march is a
// slab-clipped ray walk with paired float2 taps (volume lives in L2: 8MB).
// ---------------------------------------------------------------------------

typedef __attribute__((ext_vector_type(2))) float v2f;
typedef __attribute__((ext_vector_type(8))) float v8f;

#define SIDE 128
#define NBATCH 16
#define RAYS_PER_BATCH (SIDE * SIDE)          // 16384
#define NRAYS (NBATCH * RAYS_PER_BATCH)       // 262144
#define HALF_SM1 63.5f                        // (S-1)/2

// ---------------------------------------------------------------------------
// Preamble: base voxel coords u0 for every ray via WMMA f32 16x16x4.
//   u0[n][c] = Sum_t A[n][t] * Bm[t][c]
//   A[n] = { lin[j], lin[i], 1, 0 },  Bm = { h*R0c ; h*R1c ; h*(1-R2c) ; 0 }
// One wave handles one 16-ray tile. 8 waves / block.
// A layout (32b, 16x4): lanes 0-15 -> row m, VGPR {K0,K1}; lanes 16-31 {K2,K3}
// B layout (32b, 4x16): lanes 0-15 -> col m, VGPR {K0,K1}; lanes 16-31 {K2,K3}
// D layout (32b, 16x16): VGPR r -> M=r (lanes 0-15), M=r+8 (lanes 16-31), N=lane%16
// ---------------------------------------------------------------------------
__global__ __launch_bounds__(256) void wmma_base_kernel(
    const float* __restrict__ rotmat,   // [16,3,3]
    const float* __restrict__ coords,   // [S^3,3]; coords[t*3+2] == lin[t] exactly
    float* __restrict__ ws)             // SoA: ws[c*NRAYS + b*16384 + n]
{
  const int lane = threadIdx.x & 31;
  const int wave = (blockIdx.x << 3) + (threadIdx.x >> 5);
  const int b    = wave >> 10;          // 1024 tiles per batch
  const int tile = wave & 1023;
  const int m    = lane & 15;           // A: row M; B: col N
  const int hi   = lane >> 4;           // 0: K={0,1}, 1: K={2,3}

  // ---- A operand (no divergence before WMMA: all selects are cndmask)
  const int n = (tile << 4) + m;        // ray index within batch
  const int i = n >> 7, j = n & 127;
  const float lin_j = coords[j * 3 + 2];
  const float lin_i = coords[i * 3 + 2];
  v2f A;
  A[0] = hi ? 1.0f : lin_j;             // K2 : K0
  A[1] = hi ? 0.0f : lin_i;             // K3 : K1

  // ---- B operand (columns >= 3 are zero)
  const float* R  = rotmat + b * 9;
  const int   mc  = (m < 3) ? m : 2;    // clamped, always-in-range load
  const float msk = (m < 3) ? 1.0f : 0.0f;
  const float r0c = R[mc];              // R[0][c]
  const float r1c = R[3 + mc];          // R[1][c]
  const float r2c = R[6 + mc];          // R[2][c]
  v2f Bm;
  Bm[0] = (hi ? (HALF_SM1 * (1.0f - r2c)) : (HALF_SM1 * r0c)) * msk;
  Bm[1] = hi ? 0.0f : (HALF_SM1 * r1c) * msk;

  v8f C = {0.f, 0.f, 0.f, 0.f, 0.f, 0.f, 0.f, 0.f};
  // (neg_a, A, neg_b, B, c_mod, C, reuse_a, reuse_b)
  v8f D = __builtin_amdgcn_wmma_f32_16x16x4_f32(
      false, A, false, Bm, (short)0, C, false, false);

  // ---- scatter D columns 0..2 (lane's N = m) to SoA workspace
  if (m < 3) {
    float* plane = ws + m * NRAYS + b * RAYS_PER_BATCH + (tile << 4) + (hi << 3);
#pragma unroll
    for (int r = 0; r < 8; ++r) plane[r] = D[r];   // point M = 8*hi + r
  }
}

// ---------------------------------------------------------------------------
// Ray march: one thread per (b,i,j). u(k) = u0 + k*R2 (exact fma per step,
// no drift). Slab-clip k to the voxel cube, then branchless masked trilinear
// with x-pair float2 loads (x taps are adjacent in memory).
// ---------------------------------------------------------------------------
__global__ __launch_bounds__(256) void project_kernel(
    const float* __restrict__ rotmat,
    const float* __restrict__ vol,      // [128,128,128], z-major, x fastest
    const float* __restrict__ coords,
    const float* __restrict__ ws,
    int use_ws,
    float* __restrict__ out)            // [16,1,128,128]
{
  const int b = blockIdx.x >> 6;                       // 64 blocks per batch
  const int n = ((blockIdx.x & 63) << 8) + threadIdx.x;
  const int i = n >> 7, j = n & 127;

  const float* R = rotmat + b * 9;
  const float sx = R[6], sy = R[7], sz = R[8];         // step = R2 (unit)

  float ux, uy, uz;
  if (use_ws) {
    const int idx = b * RAYS_PER_BATCH + n;
    ux = ws[idx];
    uy = ws[NRAYS + idx];
    uz = ws[2 * NRAYS + idx];
  } else {
    const float lj = coords[j * 3 + 2], li = coords[i * 3 + 2];
    ux = HALF_SM1 * (lj * R[0] + li * R[3] - R[6] + 1.0f);
    uy = HALF_SM1 * (lj * R[1] + li * R[4] - R[7] + 1.0f);
    uz = HALF_SM1 * (lj * R[2] + li * R[5] - R[8] + 1.0f);
  }

  // ---- slab test: contribution possible only while u_c in (-1, 128)
  float kLo = -1e30f, kHi = 1e30f;
  {
    const float u0a[3] = {ux, uy, uz};
    const float sa[3]  = {sx, sy, sz};
#pragma unroll
    for (int a = 0; a < 3; ++a) {
      if (__builtin_fabsf(sa[a]) > 1e-8f) {
        const float t0 = (-1.0f - u0a[a]) / sa[a];
        const float t1 = (128.0f - u0a[a]) / sa[a];
        kLo = fmaxf(kLo, fminf(t0, t1));
        kHi = fminf(kHi, fmaxf(t0, t1));
      } else if (u0a[a] <= -1.0f || u0a[a] >= 128.0f) {
        kLo = 1.0f; kHi = 0.0f;                         // empty
      }
    }
  }
  kLo = fmaxf(kLo, 0.0f);
  kHi = fminf(kHi, 127.0f);
  int kstart = max(0,   (int)floorf(kLo) - 1);          // widen; masks make it
  int kend   = min(127, (int)ceilf(kHi) + 1);           // exact at the edges

  float acc = 0.0f;
  for (int k = kstart; k <= kend; ++k) {
    const float kf = (float)k;
    const float fx = fmaf(kf, sx, ux);
    const float fy = fmaf(kf, sy, uy);
    const float fz = fmaf(kf, sz, uz);

    const float flx = floorf(fx), fly = floorf(fy), flz = floorf(fz);
    const float w1x = fx - flx, w1y = fy - fly, w1z = fz - flz;
    const int ix = (int)flx, iy = (int)fly, iz = (int)flz;

    // zeros-padding as masked weights (clamped addresses stay in range)
    const float wx0 = ((unsigned)ix        < 128u) ? 1.0f - w1x : 0.0f;
    const float wx1 = ((unsigned)(ix + 1)  < 128u) ? w1x        : 0.0f;
    const float wy0 = ((unsigned)iy        < 128u) ? 1.0f - w1y : 0.0f;
    const float wy1 = ((unsigned)(iy + 1)  < 128u) ? w1y        : 0.0f;
    const float wz0 = ((unsigned)iz        < 128u) ? 1.0f - w1z : 0.0f;
    const float wz1 = ((unsigned)(iz + 1)  < 128u) ? w1z        : 0.0f;

    // x taps are memory-adjacent: load the pair once, route weights onto it
    const int xb = min(max(ix, 0), 126);
    const float wA = ((ix <= 126) ? wx0 : 0.0f) + ((ix < 0)  ? wx1 : 0.0f);
    const float wB = ((ix >= 127) ? wx0 : 0.0f) + ((ix >= 0) ? wx1 : 0.0f);

    const int y0 = min(max(iy, 0), 127), y1 = min(max(iy + 1, 0), 127);
    const int z0 = min(max(iz, 0), 127), z1 = min(max(iz + 1, 0), 127);

    float2 p00, p01, p10, p11;                 // 4 loads instead of 8
    __builtin_memcpy(&p00, vol + ((z0 << 14) + (y0 << 7) + xb), 8);
    __builtin_memcpy(&p01, vol + ((z0 << 14) + (y1 << 7) + xb), 8);
    __builtin_memcpy(&p10, vol + ((z1 << 14) + (y0 << 7) + xb), 8);
    __builtin_memcpy(&p11, vol + ((z1 << 14) + (y1 << 7) + xb), 8);

    const float c00 = fmaf(wB, p00.y, wA * p00.x);
    const float c01 = fmaf(wB, p01.y, wA * p01.x);
    const float c10 = fmaf(wB, p10.y, wA * p10.x);
    const float c11 = fmaf(wB, p11.y, wA * p11.x);

    const float d0 = fmaf(wy1, c01, wy0 * c00);
    const float d1 = fmaf(wy1, c11, wy0 * c10);
    acc = fmaf(wz0, d0, acc);
    acc = fmaf(wz1, d1, acc);
  }

  out[(b << 14) + n] = acc;                     // [B,1,S,S] flat: b*16384+i*128+j
}

extern "C" void kernel_launch(void* const* d_in, const int* in_sizes, int n_in,
                              void* d_out, int out_size, void* d_ws, size_t ws_size,
                              hipStream_t stream) {
  const float* rotmat = (const float*)d_in[0];   // [16,3,3]
  const float* vol    = (const float*)d_in[1];   // [128^3]
  const float* coords = (const float*)d_in[2];   // [128^3,3]
  float* out = (float*)d_out;
  float* ws  = (float*)d_ws;

  const size_t need = (size_t)3 * NRAYS * sizeof(float);   // 3 MB SoA bases
  const int use_ws = (ws != nullptr && ws_size >= need) ? 1 : 0;

  if (use_ws) {
    // 16384 WMMA tiles, 8 waves/block -> 2048 blocks
    wmma_base_kernel<<<2048, 256, 0, stream>>>(rotmat, coords, ws);
  }
  // 262144 rays, 256 threads/block -> 1024 blocks (64 per batch)
  project_kernel<<<1024, 256, 0, stream>>>(rotmat, vol, coords, ws, use_ws, out);
}